// ParameterMixture_86835648790543
// MI455X (gfx1250) — compile-verified
//
#include <hip/hip_runtime.h>

// ParameterMixture on MI455X (gfx1250):
//   weight_mixture = P(32x64) @ weight_bank(64x1M)  via V_WMMA_F32_16X16X4_F32
//   where P[b,e] = sum_k weight_probs[b,k] * (weight_indices[b,k]==e)
// Memory-bound: bank read exactly once (256MB) + streamed output (134MB).

typedef __attribute__((ext_vector_type(2))) float v2f;
typedef __attribute__((ext_vector_type(8))) float v8f;

#define NEXP   64
#define TOPK   8
#define BATCH  32
#define NDIM   (1024 * 1024)   // OUT_DIM * IN_DIM, flattened N of the GEMM
#define ODIM   1024
#define PPAD   68              // row pad: bank = (4*row + e) % 64 -> conflict-free

__global__ __launch_bounds__(256) void mixture_gemm_wmma(
    const float* __restrict__ wprobs,   // [32,8]
    const int*   __restrict__ widx,     // [32,8]
    const float* __restrict__ wbank,    // [64, NDIM] row-major
    float*       __restrict__ out)      // [32, NDIM] row-major
{
    __shared__ float sP[BATCH][PPAD];

    const int t = threadIdx.x;

    // ---- densify routing probs into P[32][64] in LDS ----
    for (int i = t; i < BATCH * PPAD; i += 256)
        ((float*)sP)[i] = 0.0f;
    __syncthreads();
    {
        // 256 threads == 32*8 (b,k) pairs; duplicate experts per row need atomics
        const int b = t >> 3;
        const int e = widx[t];
        atomicAdd(&sP[b][e], wprobs[t]);   // ds_add_f32
    }
    __syncthreads();

    // ---- WMMA GEMM: two 16-row M tiles x one 16-col N strip per wave ----
    const int lane = t & 31;
    const int wave = t >> 5;
    const int half = lane >> 4;     // 0: lanes 0-15, 1: lanes 16-31
    const int lr   = lane & 15;
    const long n0  = ((long)blockIdx.x * 8 + wave) * 16;

    v8f c0 = {};   // rows 0..15  (M tile 0)
    v8f c1 = {};   // rows 16..31 (M tile 1)

    #pragma unroll
    for (int s = 0; s < 16; ++s) {
        // A-matrix 16x4 f32 layout: VGPR0 = K = e (+2 for upper half-wave),
        // VGPR1 = K+1.  B-matrix 4x16 mirrors it (lane = N column).
        const int e = 4 * s + 2 * half;

        v2f a0 = *(const v2f*)&sP[lr][e];        // rows 0..15 of P
        v2f a1 = *(const v2f*)&sP[16 + lr][e];   // rows 16..31 of P

        const float* bp = wbank + (long)e * NDIM + n0 + lr;
        v2f bb;
        bb.x = bp[0];        // row K=e   (or e+2), col n0+lr
        bb.y = bp[NDIM];     // row K=e+1 (or e+3), col n0+lr

        c0 = __builtin_amdgcn_wmma_f32_16x16x4_f32(
                false, a0, false, bb, (short)0, c0, false, false);
        c1 = __builtin_amdgcn_wmma_f32_16x16x4_f32(
                false, a1, false, bb, (short)0, c1, false, false);
    }

    // ---- store C: VGPR r holds M = r + 8*half, N = lr  (streaming, NT) ----
    #pragma unroll
    for (int r = 0; r < 8; ++r) {
        const int m = r + 8 * half;
        __builtin_nontemporal_store(c0[r], out + (long)m        * NDIM + n0 + lr);
        __builtin_nontemporal_store(c1[r], out + (long)(16 + m) * NDIM + n0 + lr);
    }
}

__global__ __launch_bounds__(256) void mixture_bias(
    const float* __restrict__ bprobs,   // [32,8]
    const int*   __restrict__ bidx,     // [32,8]
    const float* __restrict__ bbank,    // [64,1024]
    float*       __restrict__ out)      // [32,1024]
{
    const int g = blockIdx.x * 256 + threadIdx.x;   // 0..32767
    const int b = g >> 10;
    const int o = g & 1023;
    float acc = 0.0f;
    #pragma unroll
    for (int k = 0; k < TOPK; ++k) {
        const int e = bidx[b * TOPK + k];
        acc += bprobs[b * TOPK + k] * bbank[e * ODIM + o];
    }
    out[g] = acc;
}

extern "C" void kernel_launch(void* const* d_in, const int* in_sizes, int n_in,
                              void* d_out, int out_size, void* d_ws, size_t ws_size,
                              hipStream_t stream) {
    const float* wprobs = (const float*)d_in[0];
    const float* bprobs = (const float*)d_in[1];
    const int*   widx   = (const int*)  d_in[2];
    const int*   bidx   = (const int*)  d_in[3];
    const float* wbank  = (const float*)d_in[4];
    const float* bbank  = (const float*)d_in[5];
    float* out = (float*)d_out;

    // weight_mixture: N = 1M columns, 128 columns per block (8 waves x 16)
    mixture_gemm_wmma<<<NDIM / 128, 256, 0, stream>>>(wprobs, widx, wbank, out);

    // bias_mixture appended after the 32*NDIM weight-mixture elements
    mixture_bias<<<(BATCH * ODIM) / 256, 256, 0, stream>>>(
        bprobs, bidx, bbank, out + (size_t)BATCH * NDIM);
}